// VectorQuantizer_87935160418745
// MI455X (gfx1250) — compile-verified
//
#include <hip/hip_runtime.h>

typedef __attribute__((ext_vector_type(2))) float v2f;
typedef __attribute__((ext_vector_type(4))) float f4;
typedef __attribute__((ext_vector_type(8))) float v8f;

#define DIM     256
#define KCODES  8192
#define NVEC    16384       // B*H*W = 16*32*32
#define HW      1024
#define DECAYF  0.99f
#define ONEMD   (1.0f - 0.99f)
#define BETAF   0.25f
#define EPSF    1e-5f

#define KSPLIT  8
#define TILES   (KCODES / 16 / KSPLIT)   // 64 16-code tiles per split

// ---------------- workspace layout (floats) ----------------
// zf      @ 0         : N*D   = 4194304   (z transposed to [N, D] row-major)
// wsq     @ 4194304   : K     = 8192
// counts  @ 4202496   : K     = 8192
// dw      @ 4210688   : K*D   = 2097152
// idx     @ 6307840   : N     = 16384 (int)
// scal    @ 6324224   : 2     (lossAcc, nAcc)
// pminv   @ 6324240   : N*KSPLIT = 131072
// pmini   @ 6455312   : N*KSPLIT = 131072 (int)

// ---- K0: transpose z [B, D, H, W] -> zf [N, D] (N = b*HW + h*W + w) ----
__global__ __launch_bounds__(256) void k_transpose(const float* __restrict__ z,
                                                   float* __restrict__ zf) {
  __shared__ float tile[32][33];
  const int b  = blockIdx.z;
  const int p0 = blockIdx.x * 32;   // position within H*W
  const int d0 = blockIdx.y * 32;   // channel
  const int tx = threadIdx.x, ty = threadIdx.y; // (32, 8)
#pragma unroll
  for (int i = 0; i < 4; ++i) {
    int d = d0 + ty + 8 * i;
    tile[ty + 8 * i][tx] = z[(size_t)(b * DIM + d) * HW + p0 + tx];
  }
  __syncthreads();
#pragma unroll
  for (int i = 0; i < 4; ++i) {
    int p = p0 + ty + 8 * i;
    zf[(size_t)(b * HW + p) * DIM + d0 + tx] = tile[tx][ty + 8 * i];
  }
}

// ---- K1: wsq[k] = sum_d w[k,d]^2 ; zero counts, dw, scalars ----
__global__ __launch_bounds__(256) void k_prep(const float* __restrict__ w,
                                              float* __restrict__ wsq,
                                              float* __restrict__ counts,
                                              float* __restrict__ dw,
                                              float* __restrict__ scal) {
  const int tid  = threadIdx.x;
  const int wave = tid >> 5, lane = tid & 31;
  const int k = blockIdx.x * 8 + wave;
  const f4* wr = (const f4*)(w + (size_t)k * DIM);
  f4 a = wr[lane * 2 + 0];
  f4 b = wr[lane * 2 + 1];
  float s = a.x * a.x + a.y * a.y + a.z * a.z + a.w * a.w +
            b.x * b.x + b.y * b.y + b.z * b.z + b.w * b.w;
#pragma unroll
  for (int off = 16; off >= 1; off >>= 1) s += __shfl_xor(s, off, 32);
  if (lane == 0) { wsq[k] = s; counts[k] = 0.0f; }
  // zero dw for this block's 8 rows (8*256 = 2048 floats = 512 float4)
  f4* dz = (f4*)(dw + (size_t)blockIdx.x * 8 * DIM);
  dz[tid]       = f4{0.f, 0.f, 0.f, 0.f};
  dz[tid + 256] = f4{0.f, 0.f, 0.f, 0.f};
  if (blockIdx.x == 0 && tid == 0) { scal[0] = 0.0f; scal[1] = 0.0f; }
}

// ---- K2: fused fp32-WMMA GEMM + per-row argmin, double-buffered, K-split ----
// grid = (NVEC/128) * KSPLIT blocks; block = 256 threads = 8 waves.
// Each wave owns 16 rows of z (A in registers, WMMA fp32 16x4 layout) and
// streams 1024 codes of this split in 16-code LDS tiles with double buffering:
// global loads for tile t+1 are issued before the 64-WMMA burst for tile t.
__global__ __launch_bounds__(256) void k_argmin(const float* __restrict__ zf,
                                                const float* __restrict__ w,
                                                const float* __restrict__ wsq,
                                                float* __restrict__ pminv,
                                                int* __restrict__ pmini) {
  __shared__ float wt[2][16 * 260];   // 2 x (16 codes x 256 D, padded rows)
  const int tid   = threadIdx.x;
  const int wave  = tid >> 5, lane = tid & 31;
  const int half  = lane >> 4, lc = lane & 15;
  const int split = blockIdx.x & (KSPLIT - 1);
  const int mblk  = blockIdx.x >> 3;
  const int m0    = mblk * 128 + wave * 16;
  const int mrow  = m0 + lc;
  const int k0    = split * (KCODES / KSPLIT);   // code base of this split

  const int li = tid;                // staging: thread -> float4 slots li, li+256, ...
  // Preload A (16 rows x 256 D) in WMMA fp32 16x4 A-layout:
  // lane<16 holds row lc, d = {4s, 4s+1}; lane>=16 holds row lc, d = {4s+2, 4s+3}
  v2f a[64];
  const float* zr = zf + (size_t)mrow * DIM + 2 * half;
#pragma unroll
  for (int s = 0; s < 64; ++s) a[s] = *(const v2f*)(zr + 4 * s);

  // stage tile 0 into buffer 0
  {
    const f4* src = (const f4*)(w + (size_t)k0 * DIM);
#pragma unroll
    for (int i = 0; i < 4; ++i) {
      int l = li + 256 * i;                    // 0..1023 float4 index
      *(f4*)&wt[0][(l >> 6) * 260 + (l & 63) * 4] = src[l];
    }
  }
  __syncthreads();

  float minv[8];
  int   mini[8];
#pragma unroll
  for (int r = 0; r < 8; ++r) { minv[r] = 3.4e38f; mini[r] = 0; }

  for (int t = 0; t < TILES; ++t) {
    const int buf = t & 1;

    // issue global loads for tile t+1 early (latency hidden under WMMAs)
    f4 pre0, pre1, pre2, pre3;
    if (t + 1 < TILES) {
      const f4* src = (const f4*)(w + (size_t)(k0 + (t + 1) * 16) * DIM);
      pre0 = src[li];
      pre1 = src[li + 256];
      pre2 = src[li + 512];
      pre3 = src[li + 768];
    }

    v8f acc = {};
    // B layout for 4x16 fp32: lane<16 holds rows {4s,4s+1}, lane>=16 rows
    // {4s+2,4s+3}, column (code) = lane & 15 -> one ds_load_b64 per step.
    const float* bt = &wt[buf][lc * 260 + 2 * half];
#pragma unroll
    for (int s = 0; s < 64; ++s) {
      v2f b2 = *(const v2f*)(bt + 4 * s);
      acc = __builtin_amdgcn_wmma_f32_16x16x4_f32(
          false, a[s], false, b2, (short)0, acc, false, false);
    }

    // spill prefetched tile into the alternate buffer
    if (t + 1 < TILES) {
      float* dstb = wt[buf ^ 1];
      *(f4*)&dstb[((li)       >> 6) * 260 + ((li)       & 63) * 4] = pre0;
      *(f4*)&dstb[((li + 256) >> 6) * 260 + ((li + 256) & 63) * 4] = pre1;
      *(f4*)&dstb[((li + 512) >> 6) * 260 + ((li + 512) & 63) * 4] = pre2;
      *(f4*)&dstb[((li + 768) >> 6) * 260 + ((li + 768) & 63) * 4] = pre3;
    }

    const int   col = k0 + t * 16 + lc;
    const float wq  = wsq[col];
#pragma unroll
    for (int r = 0; r < 8; ++r) {
      float score = wq - 2.0f * acc[r];
      if (score < minv[r]) { minv[r] = score; mini[r] = col; }
    }
    __syncthreads();
  }

  // reduce min/argmin across the 16 lanes sharing each row; tie -> lower index
#pragma unroll
  for (int r = 0; r < 8; ++r) {
    float v = minv[r]; int ix = mini[r];
#pragma unroll
    for (int off = 8; off >= 1; off >>= 1) {
      float ov = __shfl_xor(v, off, 32);
      int   oi = __shfl_xor(ix, off, 32);
      if (ov < v || (ov == v && oi < ix)) { v = ov; ix = oi; }
    }
    if (lc == 0) {
      int row = m0 + half * 8 + r;       // lanes 0/16 own rows r / 8+r
      pminv[(size_t)row * KSPLIT + split] = v;
      pmini[(size_t)row * KSPLIT + split] = ix;
    }
  }
}

// ---- K2b: combine per-split partial argmins (ascending split = ascending k) ----
__global__ __launch_bounds__(256) void k_combine(const float* __restrict__ pminv,
                                                 const int* __restrict__ pmini,
                                                 int* __restrict__ idxp) {
  const int n = blockIdx.x * 256 + threadIdx.x;
  float best = 3.4e38f; int bi = 0x7fffffff;
#pragma unroll
  for (int s = 0; s < KSPLIT; ++s) {
    float v = pminv[(size_t)n * KSPLIT + s];
    int  ix = pmini[(size_t)n * KSPLIT + s];
    if (v < best || (v == best && ix < bi)) { best = v; bi = ix; }
  }
  idxp[n] = bi;
}

// ---- K3: gather quantized rows, loss sum, counts, dw segment-sum ----
__global__ __launch_bounds__(256) void k_gather(const float* __restrict__ zf,
                                                const float* __restrict__ w,
                                                const int* __restrict__ idxp,
                                                float* __restrict__ outq,
                                                float* __restrict__ counts,
                                                float* __restrict__ dw,
                                                float* __restrict__ lossAcc) {
  __shared__ float sp[8];
  const int n = blockIdx.x;
  const int tid = threadIdx.x;          // = d
  const int k = idxp[n];
  float zv = zf[(size_t)n * DIM + tid];
  float wv = w[(size_t)k * DIM + tid];
  outq[(size_t)n * DIM + tid] = wv;     // quantized_st forward value == q_emb
  float diff = zv - wv;
  float s = diff * diff;
#pragma unroll
  for (int off = 16; off >= 1; off >>= 1) s += __shfl_xor(s, off, 32);
  if ((tid & 31) == 0) sp[tid >> 5] = s;
  __syncthreads();
  if (tid == 0) {
    float t = 0.0f;
#pragma unroll
    for (int i = 0; i < 8; ++i) t += sp[i];
    atomicAdd(lossAcc, t);
    atomicAdd(&counts[k], 1.0f);
  }
  atomicAdd(&dw[(size_t)k * DIM + tid], zv);
}

// ---- K4a: new_cluster_size + global sum n ----
__global__ __launch_bounds__(256) void k_ema_cs(const float* __restrict__ cs_in,
                                                const float* __restrict__ counts,
                                                float* __restrict__ cs_out,
                                                float* __restrict__ nAcc) {
  __shared__ float sp[8];
  const int k = blockIdx.x * 256 + threadIdx.x;
  float ncs = DECAYF * cs_in[k] + ONEMD * counts[k];
  cs_out[k] = ncs;
  float s = ncs;
#pragma unroll
  for (int off = 16; off >= 1; off >>= 1) s += __shfl_xor(s, off, 32);
  if ((threadIdx.x & 31) == 0) sp[threadIdx.x >> 5] = s;
  __syncthreads();
  if (threadIdx.x == 0) {
    float t = 0.0f;
#pragma unroll
    for (int i = 0; i < 8; ++i) t += sp[i];
    atomicAdd(nAcc, t);
  }
}

// ---- K4b: new_embedding_avg, new_weight, losses ----
__global__ __launch_bounds__(256) void k_finalize(const float* __restrict__ ea_in,
                                                  const float* __restrict__ dw,
                                                  const float* __restrict__ cs_out,
                                                  const float* __restrict__ nAcc,
                                                  const float* __restrict__ lossAcc,
                                                  float* __restrict__ w_out,
                                                  float* __restrict__ ea_out,
                                                  float* __restrict__ commit_out,
                                                  float* __restrict__ code_out) {
  const size_t i = (size_t)blockIdx.x * 256 + threadIdx.x;  // element index in [0, K*D)
  const int k = (int)(i >> 8);                              // / DIM
  float nea = DECAYF * ea_in[i] + ONEMD * dw[i];
  ea_out[i] = nea;
  float n   = nAcc[0];
  float ncs = cs_out[k];
  float csh = (ncs + EPSF) / (n + (float)KCODES * EPSF) * n;
  w_out[i] = nea / csh;
  if (blockIdx.x == 0 && threadIdx.x == 0) {
    float lm = lossAcc[0] / (float)((size_t)NVEC * DIM);
    code_out[0]   = lm;            // codebook_loss
    commit_out[0] = BETAF * lm;    // commitment_loss
  }
}

extern "C" void kernel_launch(void* const* d_in, const int* in_sizes, int n_in,
                              void* d_out, int out_size, void* d_ws, size_t ws_size,
                              hipStream_t stream) {
  const float* z  = (const float*)d_in[0];
  const float* w  = (const float*)d_in[1];
  const float* cs = (const float*)d_in[2];
  const float* ea = (const float*)d_in[3];
  float* out = (float*)d_out;
  float* ws  = (float*)d_ws;

  float* zf      = ws;
  float* wsq     = ws + 4194304;
  float* counts  = ws + 4202496;
  float* dw      = ws + 4210688;
  int*   idxp    = (int*)(ws + 6307840);
  float* lossAcc = ws + 6324224;   // scal[0]=loss, scal[1]=n
  float* nAcc    = ws + 6324225;
  float* pminv   = ws + 6324240;
  int*   pmini   = (int*)(ws + 6455312);

  // outputs concatenated: quantized_st, commitment_loss, codebook_loss,
  //                       new_weight, new_cluster_size, new_embedding_avg
  float* out_q      = out;
  float* out_commit = out + 4194304;
  float* out_code   = out + 4194305;
  float* out_w      = out + 4194306;
  float* out_cs     = out + 6291458;
  float* out_ea     = out + 6299650;

  k_transpose<<<dim3(32, 8, 16), dim3(32, 8), 0, stream>>>(z, zf);
  k_prep    <<<1024, 256, 0, stream>>>(w, wsq, counts, dw, lossAcc);
  k_argmin  <<<(NVEC / 128) * KSPLIT, 256, 0, stream>>>(zf, w, wsq, pminv, pmini);
  k_combine <<<NVEC / 256, 256, 0, stream>>>(pminv, pmini, idxp);
  k_gather  <<<NVEC, 256, 0, stream>>>(zf, w, idxp, out_q, counts, dw, lossAcc);
  k_ema_cs  <<<KCODES / 256, 256, 0, stream>>>(cs, counts, out_cs, nAcc);
  k_finalize<<<(KCODES * DIM) / 256, 256, 0, stream>>>(ea, dw, out_cs, nAcc, lossAcc,
                                                       out_w, out_ea, out_commit, out_code);
}